// AttentiveEncoder_17102559772923
// MI455X (gfx1250) — compile-verified
//
#include <hip/hip_runtime.h>
#include <math.h>

// ---------------- model dimensions ----------------
#define B_     4
#define HH_    16
#define WW_    16
#define C_     512
#define L_     256      // H*W
#define L3_    768      // 3*L
#define I_     1024     // 2*C
#define NS_    16
#define R_     32
#define HEADS_ 8
#define DH_    64
#define LQ_    64
#define KC3_   4608     // 9*C (im2col K)
#define BN_INV 0.9999950000374997f   // 1/sqrt(1+1e-5)

typedef __attribute__((ext_vector_type(2))) float v2f;
typedef __attribute__((ext_vector_type(8))) float v8f;

// ---------------- device helpers ----------------
__device__ __forceinline__ float wsum(float v) {
#pragma unroll
  for (int m = 16; m > 0; m >>= 1) v += __shfl_xor(v, m, 32);
  return v;
}
__device__ __forceinline__ float wmaxr(float v) {
#pragma unroll
  for (int m = 16; m > 0; m >>= 1) v = fmaxf(v, __shfl_xor(v, m, 32));
  return v;
}
__device__ __forceinline__ float siluf(float x) { return x / (1.f + expf(-x)); }

// ---------------- LDS-tiled WMMA fp32 GEMM ----------------
// C[M,N] = A[M,K] * op(B) (+bias[N]) (+C if ACCUM)
//   TB=true : B stored [N,K] (x @ W^T, dominant case)
//   TB=false: B stored [K,N]
// Block = 256 threads (8 wave32) computing a 64(M) x 32(N) tile; K stepped in
// chunks of 32 staged in LDS via coalesced float4 loads (A panel shared by 2
// waves, B panel by 4). LDS row stride padded to 36 floats: 144B rows keep
// float4 stores 16B-aligned, fragment b64 reads 8B-aligned, and the 16-row
// column walk (col + 36*r mod 64) hits 16 distinct banks (conflict-free).
// Fragments per V_WMMA_F32_16X16X4_F32:
//   A 16x4: lanes 0-15 hold M=r (v0=K0,v1=K1); lanes 16-31 (v0=K2,v1=K3)
//   B 4x16: lanes hold N=r; lo half rows K0/K1, hi half rows K2/K3
//   C/D   : v[j] = row m0+j (+8 for hi half), col n0+r
// All call sites have M%64==0, N%32==0, K%32==0 -> no tails, EXEC all-ones.
#define KB_ 32
#define LDSP_ 36
template <bool TB, bool BIAS, bool ACCUM>
__global__ void gemm_wmma(const float* __restrict__ A, long lda, long sA,
                          const float* __restrict__ Bm, long ldb, long sB,
                          float* __restrict__ Cm, long ldc, long sC,
                          const float* __restrict__ bias,
                          int M, int N, int K) {
  __shared__ float As[64][LDSP_];
  __shared__ float Bs[32][LDSP_];
  const int tid  = threadIdx.x;
  const int lane = tid & 31;
  const int wave = tid >> 5;
  const int half = lane >> 4;
  const int r    = lane & 15;
  const int m0B = blockIdx.y * 64;
  const int n0B = blockIdx.x * 32;
  const int mW  = (wave & 3) * 16;   // wave's m offset inside block tile
  const int nW  = (wave >> 2) * 16;  // wave's n offset inside block tile
  const float* Ab = A  + (long)blockIdx.z * sA;
  const float* Bb = Bm + (long)blockIdx.z * sB;

  // cooperative load coordinates: 32 rows x 8 float4-columns per pass
  const int lrow = tid >> 3;          // 0..31
  const int lcol = (tid & 7) * 4;     // 0,4,...,28

  v8f acc = {};
  for (int k0 = 0; k0 < K; k0 += KB_) {
    __syncthreads();
    // ---- stage A panel (64x32) : two float4 per thread, coalesced ----
    {
      const float4 a0 = *(const float4*)(Ab + (long)(m0B + lrow) * lda + k0 + lcol);
      const float4 a1 = *(const float4*)(Ab + (long)(m0B + 32 + lrow) * lda + k0 + lcol);
      *(float4*)(&As[lrow][lcol])      = a0;
      *(float4*)(&As[32 + lrow][lcol]) = a1;
    }
    // ---- stage B panel (32x32) : one float4 per thread, coalesced ----
    if (TB) {
      const float4 b0 = *(const float4*)(Bb + (long)(n0B + lrow) * ldb + k0 + lcol);
      *(float4*)(&Bs[lrow][lcol]) = b0;
    } else {
      const float4 b0 = *(const float4*)(Bb + (long)(k0 + lrow) * ldb + n0B + lcol);
      *(float4*)(&Bs[lrow][lcol]) = b0;
    }
    // prefetch next K-chunk while this one is consumed (global_prefetch_b8)
    if (k0 + KB_ < K) {
      __builtin_prefetch(Ab + (long)(m0B + lrow) * lda + k0 + KB_ + lcol, 0, 3);
      __builtin_prefetch(Ab + (long)(m0B + 32 + lrow) * lda + k0 + KB_ + lcol, 0, 3);
      if (TB) __builtin_prefetch(Bb + (long)(n0B + lrow) * ldb + k0 + KB_ + lcol, 0, 3);
      else    __builtin_prefetch(Bb + (long)(k0 + KB_ + lrow) * ldb + n0B + lcol, 0, 3);
    }
    __syncthreads();
    // ---- 8 WMMA steps over the staged chunk ----
#pragma unroll
    for (int kk = 0; kk < KB_; kk += 4) {
      v2f a, b;
      const float* ap = &As[mW + r][kk + 2 * half];
      a.x = ap[0]; a.y = ap[1];
      if (TB) {
        const float* bp = &Bs[nW + r][kk + 2 * half];
        b.x = bp[0]; b.y = bp[1];
      } else {
        b.x = Bs[kk + 2 * half][nW + r];
        b.y = Bs[kk + 2 * half + 1][nW + r];
      }
      acc = __builtin_amdgcn_wmma_f32_16x16x4_f32(false, a, false, b, (short)0,
                                                  acc, false, false);
    }
  }

  float* Cp = Cm + (long)blockIdx.z * sC;
  const int n = n0B + nW + r;
  const float bv = BIAS ? bias[n] : 0.f;
#pragma unroll
  for (int j = 0; j < 8; j++) {
    const int m = m0B + mW + j + 8 * half;
    const long off = (long)m * ldc + n;
    float v = acc[j] + bv;
    if (ACCUM) v += Cp[off];
    Cp[off] = v;
  }
}

// ---------------- elementwise / row kernels ----------------
__global__ void add_pe_k(const float* __restrict__ img, const float* __restrict__ h_emb,
                         const float* __restrict__ w_emb, float* __restrict__ out, long n) {
  long idx = blockIdx.x * (long)blockDim.x + threadIdx.x;
  if (idx >= n) return;
  int c = (int)(idx % C_);
  int l = (int)((idx / C_) % L_);
  int h = l >> 4, w = l & 15;
  float pe = (c < 256) ? h_emb[w * 256 + c] : w_emb[h * 256 + (c - 256)];
  out[idx] = img[idx] + pe;
}

// POOL_W(64->256) is an exact repeat-by-4: pooled[b,p,:] = emb[ques[b,p>>2],:]
__global__ void pool_gather_k(const float* __restrict__ emb, const int* __restrict__ ques,
                              float* __restrict__ out, long n) {
  long idx = blockIdx.x * (long)blockDim.x + threadIdx.x;
  if (idx >= n) return;
  int c = (int)(idx % C_);
  int p = (int)((idx / C_) % L_);
  int b = (int)(idx / ((long)C_ * L_));
  int tok = ques[b * LQ_ + (p >> 2)];
  out[idx] = emb[(long)tok * C_ + c];
}

__global__ void gemv_bias_k(const float* __restrict__ Wm, const float* __restrict__ x,
                            const float* __restrict__ bias, float* __restrict__ out,
                            int n, int k) {
  int d = blockIdx.x * blockDim.x + threadIdx.x;
  if (d >= n) return;
  const float* row = Wm + (long)d * k;
  float s = bias[d];
  for (int e = 0; e < k; e++) s += row[e] * x[e];
  out[d] = s;
}

__global__ void make_conds_k(const float* a1, const float* a2, const float* a3,
                             float* c1, float* c2, float* c3, long n) {
  long idx = blockIdx.x * (long)blockDim.x + threadIdx.x;
  if (idx >= n) return;
  float x1 = a1[idx], x2 = a2[idx], x3 = a3[idx];
  c1[idx] = x1 + x2 + x3;
  c2[idx] = 2.f * x2 + x3;   // a2+a2+a3
  c3[idx] = x2 + 2.f * x3;   // a3+a2+a3
}

__global__ void interleave3_k(const float* s1, const float* s2, const float* s3,
                              float* f, long n) {
  long idx = blockIdx.x * (long)blockDim.x + threadIdx.x;
  if (idx >= n) return;
  int c = (int)(idx % C_);
  int l = (int)((idx / C_) % L_);
  int b = (int)(idx / ((long)C_ * L_));
  long base = ((long)b * L3_ + 3 * l) * C_ + c;
  f[base] = s1[idx]; f[base + C_] = s2[idx]; f[base + 2 * C_] = s3[idx];
}

__global__ void concat3_k(const float* s1, const float* s2, const float* s3,
                          float* f, long n) {
  long idx = blockIdx.x * (long)blockDim.x + threadIdx.x;
  if (idx >= n) return;
  int c = (int)(idx % C_);
  long m = idx / C_;
  long base = m * (3 * C_) + c;
  f[base] = s1[idx]; f[base + C_] = s2[idx]; f[base + 2 * C_] = s3[idx];
}

__global__ void bn_relu_k(const float* x, const float* g, const float* b,
                          float* out, long n) {
  long idx = blockIdx.x * (long)blockDim.x + threadIdx.x;
  if (idx >= n) return;
  int c = (int)(idx % C_);
  out[idx] = fmaxf(x[idx] * (g[c] * BN_INV) + b[c], 0.f);
}

__global__ void bn_add_relu_k(const float* x, const float* g, const float* b,
                              const float* res, float* out, long n) {
  long idx = blockIdx.x * (long)blockDim.x + threadIdx.x;
  if (idx >= n) return;
  int c = (int)(idx % C_);
  out[idx] = fmaxf(x[idx] * (g[c] * BN_INV) + b[c] + res[idx], 0.f);
}

__global__ void im2col3x3_k(const float* __restrict__ x, float* __restrict__ col, long n) {
  long idx = blockIdx.x * (long)blockDim.x + threadIdx.x;
  if (idx >= n) return;
  int c = (int)(idx % C_);
  int l = (int)((idx / C_) % L_);
  int b = (int)(idx / ((long)C_ * L_));
  int h = l >> 4, w = l & 15;
  long obase = ((long)(b * L_ + l)) * KC3_ + (long)c * 9;
#pragma unroll
  for (int dy = 0; dy < 3; dy++)
#pragma unroll
    for (int dx = 0; dx < 3; dx++) {
      int hh = h + dy - 1, ww = w + dx - 1;
      float v = 0.f;
      if (hh >= 0 && hh < HH_ && ww >= 0 && ww < WW_)
        v = x[((long)(b * L_) + hh * WW_ + ww) * C_ + c];
      col[obase + dy * 3 + dx] = v;
    }
}

__global__ void conv1d_silu_k(const float* __restrict__ hg, const float* __restrict__ cw,
                              const float* __restrict__ cb, float* __restrict__ out,
                              int Ln, long n) {
  long idx = blockIdx.x * (long)blockDim.x + threadIdx.x;
  if (idx >= n) return;
  int i = (int)(idx % I_);
  int l = (int)((idx / I_) % Ln);
  int b = (int)(idx / ((long)I_ * Ln));
  const float* hb = hg + ((long)b * Ln) * (2 * I_) + i;  // h lives in cols [0,I) of hg
  float a = cb[i];
  if (l >= 2) a += hb[(long)(l - 2) * (2 * I_)] * cw[i * 3 + 0];
  if (l >= 1) a += hb[(long)(l - 1) * (2 * I_)] * cw[i * 3 + 1];
  a += hb[(long)l * (2 * I_)] * cw[i * 3 + 2];
  out[idx] = siluf(a);
}

__global__ void softplus_k(float* x, long n) {
  long idx = blockIdx.x * (long)blockDim.x + threadIdx.x;
  if (idx >= n) return;
  float v = x[idx];
  x[idx] = (v > 20.f) ? v : log1pf(expf(v));
}

__global__ void gate_silu_k(float* __restrict__ y, const float* __restrict__ hg, long n) {
  long idx = blockIdx.x * (long)blockDim.x + threadIdx.x;
  if (idx >= n) return;
  int i = (int)(idx % I_);
  long row = idx / I_;
  float g = hg[row * (2 * I_) + I_ + i];
  y[idx] *= siluf(g);
}

__global__ void add2_k(const float* a, const float* b, float* o, long n) {
  long idx = blockIdx.x * (long)blockDim.x + threadIdx.x;
  if (idx >= n) return;
  o[idx] = a[idx] + b[idx];
}

// selective scan: thread per (b, i) channel; B/C coeffs staged in LDS per step
__global__ void mamba_scan_k(const float* __restrict__ dt, const float* __restrict__ sp,
                             const float* __restrict__ h, const float* __restrict__ A_log,
                             const float* __restrict__ Dv, float* __restrict__ y, int Ln) {
  int b = blockIdx.x >> 2;                   // I_/256 = 4 blocks per batch
  int i = ((blockIdx.x & 3) << 8) + threadIdx.x;
  __shared__ float sB[NS_], sC[NS_];
  float st[NS_], Ar[NS_];
#pragma unroll
  for (int n = 0; n < NS_; n++) { st[n] = 0.f; Ar[n] = -expf(A_log[i * NS_ + n]); }
  float dcoef = Dv[i];
  for (int l = 0; l < Ln; l++) {
    long row = (long)(b * Ln + l);
    __syncthreads();
    if (threadIdx.x < NS_)           sB[threadIdx.x]       = sp[row * 64 + R_ + threadIdx.x];
    else if (threadIdx.x < 2 * NS_)  sC[threadIdx.x - NS_] = sp[row * 64 + R_ + NS_ + (threadIdx.x - NS_)];
    __syncthreads();
    float dtv = dt[row * I_ + i];
    float hv  = h[row * I_ + i];
    float acc = 0.f;
#pragma unroll
    for (int n = 0; n < NS_; n++) {
      float da = expf(dtv * Ar[n]);
      st[n] = st[n] * da + dtv * sB[n] * hv;
      acc += st[n] * sC[n];
    }
    y[row * I_ + i] = acc + hv * dcoef;
  }
}

// wave-per-row softmax over 256 columns (raw scores scaled by scale=1/sqrt(DH) on load)
__global__ void softmax256_k(float* s, int rows, float scale) {
  int row  = (int)((blockIdx.x * (long)blockDim.x + threadIdx.x) >> 5);
  int lane = threadIdx.x & 31;
  if (row >= rows) return;
  float* p = s + (long)row * 256;
  float v[8]; float mx = -1e30f;
#pragma unroll
  for (int j = 0; j < 8; j++) { v[j] = p[lane + 32 * j] * scale; mx = fmaxf(mx, v[j]); }
  mx = wmaxr(mx);
  float sum = 0.f;
#pragma unroll
  for (int j = 0; j < 8; j++) { v[j] = expf(v[j] - mx); sum += v[j]; }
  sum = wsum(sum);
  float inv = 1.f / sum;
#pragma unroll
  for (int j = 0; j < 8; j++) p[lane + 32 * j] = v[j] * inv;
}

// wave-per-row rmsnorm over C=512 (16 elems/lane)
__global__ void rmsnorm512_k(const float* x, long xs, const float* w,
                             float* out, long os, int rows) {
  int row  = (int)((blockIdx.x * (long)blockDim.x + threadIdx.x) >> 5);
  int lane = threadIdx.x & 31;
  if (row >= rows) return;
  const float* xr = x + (long)row * xs;
  float v[16]; float ss = 0.f;
#pragma unroll
  for (int j = 0; j < 16; j++) { v[j] = xr[lane + 32 * j]; ss += v[j] * v[j]; }
  ss = wsum(ss);
  float inv = rsqrtf(ss * (1.f / C_) + 1e-6f);
  float* o = out + (long)row * os;
#pragma unroll
  for (int j = 0; j < 16; j++) { int c = lane + 32 * j; o[c] = v[j] * inv * w[c]; }
}

// wave-per-row layernorm; optional "+ resid*alpha" (resid stride == out stride)
__global__ void layernorm512_k(const float* x, long xs, const float* g, const float* bt,
                               float* out, long os, const float* resid,
                               const float* alpha_p, int rows) {
  int row  = (int)((blockIdx.x * (long)blockDim.x + threadIdx.x) >> 5);
  int lane = threadIdx.x & 31;
  if (row >= rows) return;
  const float* xr = x + (long)row * xs;
  float v[16]; float s = 0.f;
#pragma unroll
  for (int j = 0; j < 16; j++) { v[j] = xr[lane + 32 * j]; s += v[j]; }
  s = wsum(s);
  float m = s * (1.f / C_);
  float q = 0.f;
#pragma unroll
  for (int j = 0; j < 16; j++) { float d = v[j] - m; q += d * d; }
  q = wsum(q);
  float inv = rsqrtf(q * (1.f / C_) + 1e-5f);
  float alpha = (resid != nullptr) ? alpha_p[0] : 0.f;
  float* o = out + (long)row * os;
#pragma unroll
  for (int j = 0; j < 16; j++) {
    int c = lane + 32 * j;
    float r = (v[j] - m) * inv * g[c] + bt[c];
    if (resid) r += resid[(long)row * os + c] * alpha;
    o[c] = r;
  }
}

// final layernorm + (B,L,C) -> (B,C,L) transpose into d_out
__global__ void layernorm_tr_k(const float* x, const float* g, const float* bt,
                               float* out, int rows) {
  int row  = (int)((blockIdx.x * (long)blockDim.x + threadIdx.x) >> 5);
  int lane = threadIdx.x & 31;
  if (row >= rows) return;
  const float* xr = x + (long)row * C_;
  float v[16]; float s = 0.f;
#pragma unroll
  for (int j = 0; j < 16; j++) { v[j] = xr[lane + 32 * j]; s += v[j]; }
  s = wsum(s);
  float m = s * (1.f / C_);
  float q = 0.f;
#pragma unroll
  for (int j = 0; j < 16; j++) { float d = v[j] - m; q += d * d; }
  q = wsum(q);
  float inv = rsqrtf(q * (1.f / C_) + 1e-5f);
  int b = row >> 8, l = row & 255;
  float* ob = out + (long)b * C_ * L_ + l;
#pragma unroll
  for (int j = 0; j < 16; j++) {
    int c = lane + 32 * j;
    ob[(long)c * L_] = (v[j] - m) * inv * g[c] + bt[c];
  }
}

// ---------------- host-side orchestration ----------------
static inline dim3 ewg(long n) { return dim3((unsigned)((n + 255) / 256)); }
static inline dim3 rowg(long rows) { return dim3((unsigned)((rows * 32 + 255) / 256)); }

static void gemm(hipStream_t st, bool tb, bool bias, bool accum,
                 const float* A, long lda, long sA,
                 const float* Bp, long ldb, long sB,
                 float* Cp, long ldc, long sC,
                 const float* bv, int M, int N, int K, int batches) {
  dim3 g((N + 31) / 32, (M + 63) / 64, batches), bl(256);
  if (tb) {
    if (accum)     gemm_wmma<true, false, true ><<<g, bl, 0, st>>>(A, lda, sA, Bp, ldb, sB, Cp, ldc, sC, bv, M, N, K);
    else if (bias) gemm_wmma<true, true,  false><<<g, bl, 0, st>>>(A, lda, sA, Bp, ldb, sB, Cp, ldc, sC, bv, M, N, K);
    else           gemm_wmma<true, false, false><<<g, bl, 0, st>>>(A, lda, sA, Bp, ldb, sB, Cp, ldc, sC, bv, M, N, K);
  } else {
    if (bias)      gemm_wmma<false, true, false><<<g, bl, 0, st>>>(A, lda, sA, Bp, ldb, sB, Cp, ldc, sC, bv, M, N, K);
    else           gemm_wmma<false, false,false><<<g, bl, 0, st>>>(A, lda, sA, Bp, ldb, sB, Cp, ldc, sC, bv, M, N, K);
  }
}

// ---- input leaf indexing (JAX pytree alphabetical flatten of params, after
// the four top-level inputs in setup_inputs insertion order) ----
enum {
  IN_IMG_A = 0, IN_IMG_B, IN_IMG_C, IN_QUES,
  P_EMB, P_IN_B, P_IN_W, P_KB, P_KW, P_OUT_B, P_OUT_W, P_QB, P_QW, P_VB, P_VW,
  P_ALPHA, P_FUSE_CONV_B, P_FUSE_CONV_W, P_H_EMB,
  P_LAYER_BASE,          // 19; each layer = ca0(12) + ca1(12) + fuse(11) = 35 leaves
};
#define LAYER_STRIDE 35
#define AFTER_LAYERS (P_LAYER_BASE + 2 * LAYER_STRIDE)   // 89
enum {
  P_LN_B = AFTER_LAYERS, P_LN_W,
  P_RB_B1, P_RB_B2, P_RB_B3, P_RB_C1B, P_RB_C1W, P_RB_C3B, P_RB_C3W,
  P_RB_G1, P_RB_G2, P_RB_G3, P_W_EMB,
};

struct Cam {
  const float *A_log, *D, *cond_w, *conv_b, *conv_w, *dt_b, *dt_w, *in_w, *out_w, *x_w;
  const float *norm_f, *norm;
};
static Cam getCam(void* const* din, int base, bool cond) {
  Cam c; int i = base;
  c.A_log = (const float*)din[i++];
  c.D     = (const float*)din[i++];
  c.cond_w = cond ? (const float*)din[i++] : nullptr;
  c.conv_b = (const float*)din[i++];
  c.conv_w = (const float*)din[i++];
  c.dt_b   = (const float*)din[i++];
  c.dt_w   = (const float*)din[i++];
  c.in_w   = (const float*)din[i++];
  c.out_w  = (const float*)din[i++];
  c.x_w    = (const float*)din[i++];
  c.norm_f = (const float*)din[i++];
  c.norm   = (const float*)din[i++];
  return c;
}

struct WS {
  float *sa1, *sa2, *sa3, *pooled, *Wq, *Wk, *Wv, *bq, *bk, *bv, *k2, *v2, *q2,
        *attb, *a1, *a2, *a3, *c1, *c2, *c3, *xn, *hg, *hconv, *sp, *dt, *y,
        *mix, *hres, *fuse, *feat3, *x0, *t1, *t2, *t3, *scores, *col;
};

static void run_alignment(hipStream_t st, WS& w, const float* sa, float* aout,
                          const float* out_w, const float* out_b) {
  const int M = B_ * L_;
  // q (two fused projections) — K/V were computed once outside
  gemm(st, true, true, false, sa, C_, 0, w.Wq, C_, 0, w.q2, C_, 0, w.bq, M, C_, C_, 1);
  // scores[b,h] = q_h @ k_h^T  (batched over b, looped over heads)
  for (int h = 0; h < HEADS_; h++)
    gemm(st, true, false, false,
         w.q2 + h * DH_, C_, (long)L_ * C_,
         w.k2 + h * DH_, C_, (long)L_ * C_,
         w.scores + (long)h * L_ * L_, L_, (long)HEADS_ * L_ * L_,
         nullptr, L_, L_, DH_, B_);
  softmax256_k<<<rowg(B_ * HEADS_ * L_), 256, 0, st>>>(w.scores, B_ * HEADS_ * L_, 0.125f);
  // out[b,h] = att @ v_h  (B stored [K,N] strided inside v2)
  for (int h = 0; h < HEADS_; h++)
    gemm(st, false, false, false,
         w.scores + (long)h * L_ * L_, L_, (long)HEADS_ * L_ * L_,
         w.v2 + h * DH_, C_, (long)L_ * C_,
         w.attb + h * DH_, C_, (long)L_ * C_,
         nullptr, L_, DH_, L_, B_);
  gemm(st, true, true, false, w.attb, C_, 0, out_w, C_, 0, aout, C_, 0, out_b, M, C_, C_, 1);
}

static void run_camamba(hipStream_t st, const Cam& p, float* x, const float* cond,
                        int Ln, WS& w) {
  const int M = B_ * Ln;
  rmsnorm512_k<<<rowg(M), 256, 0, st>>>(x, C_, p.norm, w.xn, C_, M);
  gemm(st, true, false, false, w.xn, C_, 0, p.in_w, C_, 0, w.hg, 2 * I_, 0,
       nullptr, M, 2 * I_, C_, 1);
  if (cond)
    gemm(st, true, false, true, cond, C_, 0, p.cond_w, C_, 0, w.hg, 2 * I_, 0,
         nullptr, M, I_, C_, 1);
  long nMI = (long)M * I_;
  conv1d_silu_k<<<ewg(nMI), 256, 0, st>>>(w.hg, p.conv_w, p.conv_b, w.hconv, Ln, nMI);
  gemm(st, true, false, false, w.hconv, I_, 0, p.x_w, I_, 0, w.sp, 64, 0,
       nullptr, M, 64, I_, 1);
  gemm(st, true, true, false, w.sp, 64, 0, p.dt_w, R_, 0, w.dt, I_, 0,
       p.dt_b, M, I_, R_, 1);
  softplus_k<<<ewg(nMI), 256, 0, st>>>(w.dt, nMI);
  mamba_scan_k<<<dim3(B_ * (I_ / 256)), 256, 0, st>>>(w.dt, w.sp, w.hconv,
                                                      p.A_log, p.D, w.y, Ln);
  gate_silu_k<<<ewg(nMI), 256, 0, st>>>(w.y, w.hg, nMI);
  gemm(st, true, false, false, w.y, I_, 0, p.out_w, I_, 0, w.mix, C_, 0,
       nullptr, M, C_, I_, 1);
  long nMC = (long)M * C_;
  add2_k<<<ewg(nMC), 256, 0, st>>>(x, w.mix, w.hres, nMC);
  rmsnorm512_k<<<rowg(M), 256, 0, st>>>(w.hres, C_, p.norm_f, x, C_, M);
}

extern "C" void kernel_launch(void* const* d_in, const int* in_sizes, int n_in,
                              void* d_out, int out_size, void* d_ws, size_t ws_size,
                              hipStream_t stream) {
  (void)in_sizes; (void)n_in; (void)out_size; (void)ws_size;
  auto F = [&](int i) { return (const float*)d_in[i]; };

  const float* imgA = F(IN_IMG_A); const float* imgB = F(IN_IMG_B);
  const float* imgC = F(IN_IMG_C);
  const int*   ques = (const int*)d_in[IN_QUES];
  const float* emb  = F(P_EMB);
  const float* in_b = F(P_IN_B);  const float* in_w = F(P_IN_W);
  const float* kb = F(P_KB);      const float* kw = F(P_KW);
  const float* out_b_al = F(P_OUT_B); const float* out_w_al = F(P_OUT_W);
  const float* qb = F(P_QB);      const float* qw = F(P_QW);
  const float* vb = F(P_VB);      const float* vw = F(P_VW);
  const float* alpha = F(P_ALPHA);
  const float* fcb = F(P_FUSE_CONV_B); const float* fcw = F(P_FUSE_CONV_W);
  const float* h_emb = F(P_H_EMB);     const float* w_emb = F(P_W_EMB);
  const float* ln_b = F(P_LN_B);       const float* ln_w = F(P_LN_W);

  // workspace bump allocator (float units, 256B aligned chunks)
  float* base = (float*)d_ws;
  size_t off = 0;
  auto alloc = [&](size_t n) { float* p = base + off; off += (n + 63) & ~(size_t)63; return p; };
  const size_t BLC = (size_t)B_ * L_ * C_;
  WS w;
  w.sa1 = alloc(BLC); w.sa2 = alloc(BLC); w.sa3 = alloc(BLC);
  w.pooled = alloc(BLC);
  w.Wq = alloc((size_t)C_ * C_); w.Wk = alloc((size_t)C_ * C_); w.Wv = alloc((size_t)C_ * C_);
  w.bq = alloc(C_); w.bk = alloc(C_); w.bv = alloc(C_);
  w.k2 = alloc(BLC); w.v2 = alloc(BLC); w.q2 = alloc(BLC); w.attb = alloc(BLC);
  w.a1 = alloc(BLC); w.a2 = alloc(BLC); w.a3 = alloc(BLC);
  w.c1 = alloc(BLC); w.c2 = alloc(BLC); w.c3 = alloc(BLC);
  w.xn = alloc((size_t)B_ * L3_ * C_);
  w.hg = alloc((size_t)B_ * L3_ * 2 * I_);     // 6.29M floats; aliased below
  w.hconv = alloc((size_t)B_ * L3_ * I_);
  w.sp = alloc((size_t)B_ * L3_ * 64);
  w.dt = alloc((size_t)B_ * L3_ * I_);
  w.y  = alloc((size_t)B_ * L3_ * I_);
  w.mix = alloc((size_t)B_ * L3_ * C_);
  w.hres = alloc((size_t)B_ * L3_ * C_);
  w.fuse = alloc((size_t)B_ * L3_ * C_);
  w.feat3 = alloc((size_t)B_ * L_ * 3 * C_);
  w.x0 = alloc(BLC); w.t1 = alloc(BLC); w.t2 = alloc(BLC); w.t3 = alloc(BLC);
  w.scores = w.hg;   // attention scores (2.1M) alias hg (disjoint lifetime)
  w.col    = w.hg;   // im2col buffer (4.7M) alias hg (disjoint lifetime)

  const long nBLC = (long)BLC;
  const int rowsBL = B_ * L_;

  // positional embeddings + pooled question embedding (K/V source, layer-invariant)
  add_pe_k<<<ewg(nBLC), 256, 0, stream>>>(imgA, h_emb, w_emb, w.sa1, nBLC);
  add_pe_k<<<ewg(nBLC), 256, 0, stream>>>(imgB, h_emb, w_emb, w.sa2, nBLC);
  add_pe_k<<<ewg(nBLC), 256, 0, stream>>>(imgC, h_emb, w_emb, w.sa3, nBLC);
  pool_gather_k<<<ewg(nBLC), 256, 0, stream>>>(emb, ques, w.pooled, nBLC);

  // fuse the back-to-back projection pairs: W_eff = w{q,k,v} @ {q,k,v}w (NN gemm)
  gemm(stream, false, false, false, in_w,               C_, 0, qw, C_, 0, w.Wq, C_, 0, nullptr, C_, C_, C_, 1);
  gemm(stream, false, false, false, in_w + (long)C_*C_, C_, 0, kw, C_, 0, w.Wk, C_, 0, nullptr, C_, C_, C_, 1);
  gemm(stream, false, false, false, in_w + 2L*C_*C_,    C_, 0, vw, C_, 0, w.Wv, C_, 0, nullptr, C_, C_, C_, 1);
  gemv_bias_k<<<dim3(2), 256, 0, stream>>>(in_w,               qb, in_b,         w.bq, C_, C_);
  gemv_bias_k<<<dim3(2), 256, 0, stream>>>(in_w + (long)C_*C_, kb, in_b + C_,    w.bk, C_, C_);
  gemv_bias_k<<<dim3(2), 256, 0, stream>>>(in_w + 2L*C_*C_,    vb, in_b + 2*C_,  w.bv, C_, C_);

  // K/V depend only on Ques -> compute once for all layers/images
  gemm(stream, true, true, false, w.pooled, C_, 0, w.Wk, C_, 0, w.k2, C_, 0, w.bk, rowsBL, C_, C_, 1);
  gemm(stream, true, true, false, w.pooled, C_, 0, w.Wv, C_, 0, w.v2, C_, 0, w.bv, rowsBL, C_, C_, 1);

  for (int li = 0; li < 2; li++) {
    Cam ca0 = getCam(d_in, P_LAYER_BASE + LAYER_STRIDE * li + 0,  true);
    Cam ca1 = getCam(d_in, P_LAYER_BASE + LAYER_STRIDE * li + 12, true);
    Cam cfz = getCam(d_in, P_LAYER_BASE + LAYER_STRIDE * li + 24, false);

    run_alignment(stream, w, w.sa1, w.a1, out_w_al, out_b_al);
    run_alignment(stream, w, w.sa2, w.a2, out_w_al, out_b_al);
    run_alignment(stream, w, w.sa3, w.a3, out_w_al, out_b_al);
    make_conds_k<<<ewg(nBLC), 256, 0, stream>>>(w.a1, w.a2, w.a3, w.c1, w.c2, w.c3, nBLC);

    run_camamba(stream, ca0, w.sa1, w.c1, L_, w);
    run_camamba(stream, ca1, w.sa2, w.c2, L_, w);   // images 2 & 3 share ca1 params
    run_camamba(stream, ca1, w.sa3, w.c3, L_, w);

    // mid-layer layernorm (in-place; rows fully staged in registers)
    layernorm512_k<<<rowg(rowsBL), 256, 0, stream>>>(w.sa1, C_, ln_w, ln_b, w.sa1, C_, nullptr, nullptr, rowsBL);
    layernorm512_k<<<rowg(rowsBL), 256, 0, stream>>>(w.sa2, C_, ln_w, ln_b, w.sa2, C_, nullptr, nullptr, rowsBL);
    layernorm512_k<<<rowg(rowsBL), 256, 0, stream>>>(w.sa3, C_, ln_w, ln_b, w.sa3, C_, nullptr, nullptr, rowsBL);

    // fuse pass over interleaved (B, 3L, C)
    interleave3_k<<<ewg(nBLC), 256, 0, stream>>>(w.sa1, w.sa2, w.sa3, w.fuse, nBLC);
    run_camamba(stream, cfz, w.fuse, nullptr, L3_, w);

    // unstack + ln + residual*alpha : row (b,l) of image j lives at fuse + j*C, stride 3C
    layernorm512_k<<<rowg(rowsBL), 256, 0, stream>>>(w.fuse + 0 * C_, 3 * C_, ln_w, ln_b, w.sa1, C_, w.sa1, alpha, rowsBL);
    layernorm512_k<<<rowg(rowsBL), 256, 0, stream>>>(w.fuse + 1 * C_, 3 * C_, ln_w, ln_b, w.sa2, C_, w.sa2, alpha, rowsBL);
    layernorm512_k<<<rowg(rowsBL), 256, 0, stream>>>(w.fuse + 2 * C_, 3 * C_, ln_w, ln_b, w.sa3, C_, w.sa3, alpha, rowsBL);
  }

  // conv head, token-major (channels-last): 1x1 convs are plain gemms,
  // 3x3 conv is im2col (K = c*9 + dy*3 + dx, matching OIHW weight flattening)
  concat3_k<<<ewg(nBLC), 256, 0, stream>>>(w.sa1, w.sa2, w.sa3, w.feat3, nBLC);
  gemm(stream, true, true, false, w.feat3, 3 * C_, 0, fcw, 3 * C_, 0, w.x0, C_, 0,
       fcb, rowsBL, C_, 3 * C_, 1);
  bn_relu_k<<<ewg(nBLC), 256, 0, stream>>>(w.x0, F(P_RB_G1), F(P_RB_B1), w.t1, nBLC);
  im2col3x3_k<<<ewg(nBLC), 256, 0, stream>>>(w.t1, w.col, nBLC);
  gemm(stream, true, true, false, w.col, KC3_, 0, F(P_RB_C3W), KC3_, 0, w.t2, C_, 0,
       F(P_RB_C3B), rowsBL, C_, KC3_, 1);
  bn_relu_k<<<ewg(nBLC), 256, 0, stream>>>(w.t2, F(P_RB_G2), F(P_RB_B2), w.t2, nBLC);
  gemm(stream, true, true, false, w.t2, C_, 0, F(P_RB_C1W), C_, 0, w.t3, C_, 0,
       F(P_RB_C1B), rowsBL, C_, C_, 1);
  bn_add_relu_k<<<ewg(nBLC), 256, 0, stream>>>(w.t3, F(P_RB_G3), F(P_RB_B3), w.x0, w.t3, nBLC);

  layernorm_tr_k<<<rowg(rowsBL), 256, 0, stream>>>(w.t3, ln_w, ln_b, (float*)d_out, rowsBL);
}